// SineGraphConvBlock_39754217292301
// MI455X (gfx1250) — compile-verified
//
#include <hip/hip_runtime.h>
#include <hip/hip_bf16.h>

typedef float v2f __attribute__((ext_vector_type(2)));
typedef float v8f __attribute__((ext_vector_type(8)));

#define NNODES 20000
#define NEDGES 320000

// ---------------------------------------------------------------------------
// Generic fp32 GEMM:  C[M,Nc] = act(A[M,K] @ W[K,Nc] + bias)
// One wave computes a 16x16 tile via V_WMMA_F32_16X16X4_F32.
// Block = 128 threads (4 waves) -> 16 rows x 64 cols per block.
// grid = (M/16, Nc/64).  Requires M%16==0, K%4==0, Nc%64==0.
// ACT: 0=none, 1=tanh, 2=relu
// ---------------------------------------------------------------------------
template <int ACT>
__global__ __launch_bounds__(128) void gemm_kernel(const float* __restrict__ A,
                                                   const float* __restrict__ W,
                                                   const float* __restrict__ bias,
                                                   float* __restrict__ C,
                                                   int M, int K, int Nc) {
  const int lane = threadIdx.x & 31;
  const int wave = threadIdx.x >> 5;
  const int hl   = lane >> 4;    // 0: K pair {0,1}, 1: K pair {2,3}
  const int l15  = lane & 15;
  const int row0 = blockIdx.x * 16;
  const int col  = blockIdx.y * 64 + wave * 16 + l15;

  const float* arow = A + (size_t)(row0 + l15) * K;

  v8f acc = {};
  for (int k = 0; k < K; k += 4) {
    const int kk = k + 2 * hl;
    v2f a = *(const v2f*)(arow + kk);           // contiguous 8B load
    v2f b;
    b.x = W[(size_t)kk * Nc + col];
    b.y = W[(size_t)(kk + 1) * Nc + col];
    acc = __builtin_amdgcn_wmma_f32_16x16x4_f32(false, a, false, b,
                                                (short)0, acc, false, false);
  }

  const float bv = bias ? bias[col] : 0.0f;
#pragma unroll
  for (int r = 0; r < 8; ++r) {
    float v = acc[r] + bv;
    if (ACT == 1) v = tanhf(v);
    if (ACT == 2) v = fmaxf(v, 0.0f);
    C[(size_t)(row0 + r + 8 * hl) * Nc + col] = v;
  }
}

// ---------------------------------------------------------------------------
// Fused edge kernel. One block = 16 edges x DOUT outputs (DOUT/16 waves).
//   h        = relu(pre_i[dst] + pre_j[src] + b1)        [16 x 64] (LDS)
//   coupling = sigmoid(h @ W2 + b2)                      [16 x DOUT] (WMMA)
//   msg      = coupling * sin(msg_ph[src] - phases[dst])
//   atomicAdd into s[dst]
// LDS tile padded to stride 68 floats -> conflict-free b64 loads.
// ---------------------------------------------------------------------------
template <int DOUT>
__global__ __launch_bounds__(DOUT / 16 * 32) void edge_kernel(
    const long long* __restrict__ src, const long long* __restrict__ dst,
    const float* __restrict__ pre_i, const float* __restrict__ pre_j,
    const float* __restrict__ b1, const float* __restrict__ W2,
    const float* __restrict__ b2, const float* __restrict__ msgph,
    const float* __restrict__ phases, float* __restrict__ sbuf) {
  constexpr int NT = DOUT / 16 * 32;
  __shared__ float hs[16][68];
  __shared__ int sS[16], sD[16];

  const int tid = threadIdx.x;
  const int e0  = blockIdx.x * 16;
  if (tid < 16) {
    sS[tid] = (int)src[e0 + tid];
    sD[tid] = (int)dst[e0 + tid];
  }
  __syncthreads();

  // Build ReLU'd h tile (16 edges x 64), coalesced per-row gathers.
#pragma unroll
  for (int idx = tid; idx < 16 * 64; idx += NT) {
    const int er = idx >> 6;
    const int c  = idx & 63;
    float v = pre_i[(size_t)sD[er] * 64 + c] + pre_j[(size_t)sS[er] * 64 + c] + b1[c];
    hs[er][c] = fmaxf(v, 0.0f);
  }
  __syncthreads();

  const int lane = tid & 31;
  const int wave = tid >> 5;
  const int hl   = lane >> 4;
  const int l15  = lane & 15;
  const int col  = wave * 16 + l15;

  v8f acc = {};
#pragma unroll
  for (int k = 0; k < 64; k += 4) {
    const int kk = k + 2 * hl;
    v2f a = *(const v2f*)(&hs[l15][kk]);        // ds_load_b64, conflict-free
    v2f b;
    b.x = W2[(size_t)kk * DOUT + col];
    b.y = W2[(size_t)(kk + 1) * DOUT + col];
    acc = __builtin_amdgcn_wmma_f32_16x16x4_f32(false, a, false, b,
                                                (short)0, acc, false, false);
  }

  const float bb = b2[col];
#pragma unroll
  for (int r = 0; r < 8; ++r) {
    const int er = r + 8 * hl;
    const int sn = sS[er], dn = sD[er];
    const float coup = 1.0f / (1.0f + __expf(-(acc[r] + bb)));
    const float m =
        coup * __sinf(msgph[(size_t)sn * DOUT + col] - phases[(size_t)dn * DOUT + col]);
    atomicAdd(&sbuf[(size_t)dn * DOUT + col], m);
  }
}

// ---------------------------------------------------------------------------
// Degree count (float) via global atomics.
// ---------------------------------------------------------------------------
__global__ void count_kernel(const long long* __restrict__ dst, float* __restrict__ cnt,
                             int E) {
  const int i = blockIdx.x * 256 + threadIdx.x;
  if (i < E) atomicAdd(&cnt[(int)dst[i]], 1.0f);
}

// agg = s / max(cnt, 1)
__global__ void scale_rows_kernel(const float* __restrict__ s, const float* __restrict__ cnt,
                                  float* __restrict__ agg, int N, int C) {
  const int i = blockIdx.x * 256 + threadIdx.x;
  if (i < N * C) {
    const int n = i / C;
    agg[i] = s[i] / fmaxf(cnt[n], 1.0f);
  }
}

// relu(layernorm(row128)) — one wave32 per node, float4 per lane, shfl reduce.
__global__ __launch_bounds__(256) void ln_relu_kernel(const float* __restrict__ in,
                                                      const float* __restrict__ g,
                                                      const float* __restrict__ b,
                                                      float* __restrict__ out, int N) {
  const int wave = threadIdx.x >> 5;
  const int lane = threadIdx.x & 31;
  const int n    = blockIdx.x * 8 + wave;
  if (n >= N) return;

  float4 v = ((const float4*)(in + (size_t)n * 128))[lane];
  float s = v.x + v.y + v.z + v.w;
#pragma unroll
  for (int o = 16; o > 0; o >>= 1) s += __shfl_xor(s, o, 32);
  const float mu = s * (1.0f / 128.0f);

  const float dx = v.x - mu, dy = v.y - mu, dz = v.z - mu, dw = v.w - mu;
  float q = dx * dx + dy * dy + dz * dz + dw * dw;
#pragma unroll
  for (int o = 16; o > 0; o >>= 1) q += __shfl_xor(q, o, 32);
  const float rs = rsqrtf(q * (1.0f / 128.0f) + 1e-5f);

  const float4 gv = ((const float4*)g)[lane];
  const float4 bv = ((const float4*)b)[lane];
  float4 o4;
  o4.x = fmaxf(dx * rs * gv.x + bv.x, 0.0f);
  o4.y = fmaxf(dy * rs * gv.y + bv.y, 0.0f);
  o4.z = fmaxf(dz * rs * gv.z + bv.z, 0.0f);
  o4.w = fmaxf(dw * rs * gv.w + bv.w, 0.0f);
  ((float4*)(out + (size_t)n * 128))[lane] = o4;
}

__global__ void residual_kernel(const float* __restrict__ a, const float* __restrict__ x,
                                float* __restrict__ out, int total) {
  const int i = blockIdx.x * 256 + threadIdx.x;
  if (i < total) out[i] = a[i] + x[i];
}

// ---------------------------------------------------------------------------
static void launch_gemm(int act, const float* A, const float* W, const float* bias,
                        float* C, int M, int K, int Nc, hipStream_t st) {
  dim3 grid(M / 16, Nc / 64), blk(128);
  switch (act) {
    case 1: gemm_kernel<1><<<grid, blk, 0, st>>>(A, W, bias, C, M, K, Nc); break;
    case 2: gemm_kernel<2><<<grid, blk, 0, st>>>(A, W, bias, C, M, K, Nc); break;
    default: gemm_kernel<0><<<grid, blk, 0, st>>>(A, W, bias, C, M, K, Nc); break;
  }
}

extern "C" void kernel_launch(void* const* d_in, const int* in_sizes, int n_in,
                              void* d_out, int out_size, void* d_ws, size_t ws_size,
                              hipStream_t stream) {
  const int N = NNODES, E = NEDGES;
  const float* x = (const float*)d_in[0];
  const long long* ei = (const long long*)d_in[1];
  const long long* src = ei;
  const long long* dst = ei + E;

  int pi = 2;
  auto F = [&](void) { return (const float*)d_in[pi++]; };
  // layer0 (din=64, dout=128)
  const float *L0pW1 = F(), *L0pb1 = F(), *L0pW2 = F(), *L0pb2 = F();
  const float *L0mW1 = F(), *L0mb1 = F(), *L0mW2 = F(), *L0mb2 = F();
  const float *L0wW1 = F(), *L0wb1 = F(), *L0wW2 = F(), *L0wb2 = F();
  const float *L0oW1 = F(), *L0ob1 = F(), *L0oW2 = F(), *L0ob2 = F();
  // layer1 (din=128, dout=64)
  const float *L1pW1 = F(), *L1pb1 = F(), *L1pW2 = F(), *L1pb2 = F();
  const float *L1mW1 = F(), *L1mb1 = F(), *L1mW2 = F(), *L1mb2 = F();
  const float *L1wW1 = F(), *L1wb1 = F(), *L1wW2 = F(), *L1wb2 = F();
  const float *L1oW1 = F(), *L1ob1 = F(), *L1oW2 = F(), *L1ob2 = F();
  const float *LNg = F(), *LNb = F();

  // workspace
  float* ws     = (float*)d_ws;
  float* phases = ws;                   // N*128 (also reused as agg / final pre-residual)
  float* msgph  = phases + (size_t)N * 128;
  float* pre_i  = msgph + (size_t)N * 128;   // N*64
  float* pre_j  = pre_i + (size_t)N * 64;    // N*64
  float* sbuf   = pre_j + (size_t)N * 64;    // N*128
  float* cnt    = sbuf + (size_t)N * 128;    // N
  float* t1     = cnt + N;                   // N*64
  float* mlpout = t1 + (size_t)N * 64;       // N*128 (also reused as agg)
  float* hnode  = mlpout + (size_t)N * 128;  // N*128

  // degree counts (same dst for both layers)
  hipMemsetAsync(cnt, 0, (size_t)N * sizeof(float), stream);
  count_kernel<<<(E + 255) / 256, 256, 0, stream>>>(dst, cnt, E);

  // ---------------- layer 0 (din=64 -> dout=128) ----------------
  launch_gemm(1, x, L0pW1, L0pb1, t1, N, 64, 64, stream);
  launch_gemm(0, t1, L0pW2, L0pb2, phases, N, 64, 128, stream);
  launch_gemm(1, x, L0mW1, L0mb1, t1, N, 64, 64, stream);
  launch_gemm(0, t1, L0mW2, L0mb2, msgph, N, 64, 128, stream);
  launch_gemm(0, x, L0wW1, nullptr, pre_i, N, 64, 64, stream);            // rows 0..63
  launch_gemm(0, x, L0wW1 + 64 * 64, nullptr, pre_j, N, 64, 64, stream);  // rows 64..127

  hipMemsetAsync(sbuf, 0, (size_t)N * 128 * sizeof(float), stream);
  edge_kernel<128><<<E / 16, 256, 0, stream>>>(src, dst, pre_i, pre_j, L0wb1, L0wW2,
                                               L0wb2, msgph, phases, sbuf);

  scale_rows_kernel<<<((size_t)N * 128 + 255) / 256, 256, 0, stream>>>(sbuf, cnt, phases,
                                                                       N, 128);
  launch_gemm(2, phases, L0oW1, L0ob1, t1, N, 128, 64, stream);
  launch_gemm(0, t1, L0oW2, L0ob2, mlpout, N, 64, 128, stream);
  ln_relu_kernel<<<N / 8, 256, 0, stream>>>(mlpout, LNg, LNb, hnode, N);

  // ---------------- layer 1 (din=128 -> dout=64) ----------------
  launch_gemm(1, hnode, L1pW1, L1pb1, t1, N, 128, 64, stream);
  launch_gemm(0, t1, L1pW2, L1pb2, phases, N, 64, 64, stream);
  launch_gemm(1, hnode, L1mW1, L1mb1, t1, N, 128, 64, stream);
  launch_gemm(0, t1, L1mW2, L1mb2, msgph, N, 64, 64, stream);
  launch_gemm(0, hnode, L1wW1, nullptr, pre_i, N, 128, 64, stream);             // rows 0..127
  launch_gemm(0, hnode, L1wW1 + 128 * 64, nullptr, pre_j, N, 128, 64, stream);  // rows 128..255

  hipMemsetAsync(sbuf, 0, (size_t)N * 64 * sizeof(float), stream);
  edge_kernel<64><<<E / 16, 128, 0, stream>>>(src, dst, pre_i, pre_j, L1wb1, L1wW2,
                                              L1wb2, msgph, phases, sbuf);

  scale_rows_kernel<<<((size_t)N * 64 + 255) / 256, 256, 0, stream>>>(sbuf, cnt, mlpout,
                                                                      N, 64);
  launch_gemm(2, mlpout, L1oW1, L1ob1, t1, N, 64, 64, stream);
  launch_gemm(0, t1, L1oW2, L1ob2, phases, N, 64, 64, stream);

  residual_kernel<<<(N * 64 + 255) / 256, 256, 0, stream>>>(phases, x, (float*)d_out,
                                                            N * 64);
}